// Model_45947560133068
// MI455X (gfx1250) — compile-verified
//
#include <hip/hip_runtime.h>
#include <hip/hip_bf16.h>
#include <cmath>

typedef __attribute__((ext_vector_type(2))) float v2f;
typedef __attribute__((ext_vector_type(4))) float v4f;
typedef __attribute__((ext_vector_type(8))) float v8f;

#define B_ 128
#define T_ 512
#define I_ 512
#define H_ 1024
#define R_TOTAL (B_ * T_)   // 65536 rows for the input projection GEMM
#define NBLK 64             // persistent workgroups (one per 16-col N tile)

// ---------------------------------------------------------------------------
// init: h0 = 0, out = b_out, barrier counter = 0 (re-zeroed on every call so
// graph replays are deterministic)
// ---------------------------------------------------------------------------
__global__ void rnn_init(float* __restrict__ h0, float* __restrict__ out,
                         const float* __restrict__ b_out,
                         unsigned int* __restrict__ bar) {
    int idx = blockIdx.x * blockDim.x + threadIdx.x;
    if (idx < B_ * H_) h0[idx] = 0.0f;
    if (idx < B_ * T_) out[idx] = b_out[0];
    if (idx == 0) *bar = 0u;
}

// ---------------------------------------------------------------------------
// Phase 1: xp[r][n] = sum_k x[r][k] * W_ih[n][k] + (b_ih[n] + b_hh[n])
// Wave tile: 16(M) x 64(N), K looped in LDS-staged chunks of 64.
// Block = 8 waves = 128(M) x 64(N). Grid = 512 Mblk x 16 Nblk (N fastest).
// ---------------------------------------------------------------------------
__global__ void __launch_bounds__(256)
rnn_xproj(const float* __restrict__ x, const float* __restrict__ W_ih,
          const float* __restrict__ b_ih, const float* __restrict__ b_hh,
          float* __restrict__ xp)
{
    __shared__ float bsm[64 * 68];   // W_ih slice [n_local][k_local], pitch 68

    const int lane    = threadIdx.x & 31;
    const int wave    = threadIdx.x >> 5;
    const int nb      = blockIdx.x & 15;
    const int mb      = blockIdx.x >> 4;
    const int n0      = nb * 64;
    const int m0      = mb * 128 + wave * 16;
    const int halfsel = lane >> 4;       // 0: K pair {0,1}, 1: K pair {2,3}
    const int l16     = lane & 15;

    v8f c[4];
    #pragma unroll
    for (int j = 0; j < 4; ++j) {
        const int n = n0 + j * 16 + l16;
        const float bias = b_ih[n] + b_hh[n];
        #pragma unroll
        for (int v = 0; v < 8; ++v) c[j][v] = bias;
    }

    const float* arow = x + (size_t)(m0 + l16) * I_;

    for (int k0 = 0; k0 < I_; k0 += 64) {
        __syncthreads();
        {   // cooperative stage of W_ih[n0..n0+63][k0..k0+63]
            const int col = (threadIdx.x & 15) * 4;
            int r = threadIdx.x >> 4;               // 0..15
            #pragma unroll
            for (int i = 0; i < 4; ++i, r += 16) {
                v4f w = *(const v4f*)(W_ih + (size_t)(n0 + r) * I_ + k0 + col);
                *(v4f*)(&bsm[r * 68 + col]) = w;
            }
        }
        __syncthreads();

        #pragma unroll 4
        for (int kk = 0; kk < 64; kk += 4) {
            const int ka = kk + 2 * halfsel;
            v2f a = *(const v2f*)(arow + k0 + ka);
            #pragma unroll
            for (int j = 0; j < 4; ++j) {
                v2f b = *(const v2f*)(&bsm[(j * 16 + l16) * 68 + ka]);
                c[j] = __builtin_amdgcn_wmma_f32_16x16x4_f32(
                    false, a, false, b, (short)0, c[j], false, false);
            }
        }
    }

    // D layout: VGPR v -> row m0 + v + 8*halfsel, lane%16 -> column
    #pragma unroll
    for (int j = 0; j < 4; ++j) {
        const int n = n0 + j * 16 + l16;
        #pragma unroll
        for (int v = 0; v < 8; ++v) {
            const int m = m0 + v + 8 * halfsel;
            xp[(size_t)m * H_ + n] = c[j][v];
        }
    }
}

// ---------------------------------------------------------------------------
// Phase 2: ONE persistent kernel for the whole recurrence.
//   for t in 0..T-1:  h_next = tanh(xp[:,t,:] + h_prev @ W_hh^T)
//                     out[:,t] += h_next @ W_out^T   (shuffle + atomicAdd)
// 64 workgroups, each owns a 16-column N tile. Its W_hh slice (16x1024 f32 =
// 64 KB) is staged into LDS ONCE (k-major [k][n]: conflict-free reads, exactly
// 64 KB of the 320 KB/WGP) and reused for all 512 steps. Cross-step ordering
// via a device-scope split barrier: release fence + atomic arrive, spin on a
// monotonically increasing target with s_sleep, acquire fence.
// ---------------------------------------------------------------------------
__global__ void __launch_bounds__(256)
rnn_recur(const float* __restrict__ xp, const float* __restrict__ W_hh,
          const float* __restrict__ W_out,
          float* __restrict__ hA, float* __restrict__ hB,
          float* __restrict__ out, unsigned int* __restrict__ bar)
{
    __shared__ float bsm[H_ * 16];   // [k][n_local], 64 KB, bank-conflict free

    const int lane    = threadIdx.x & 31;
    const int wave    = threadIdx.x >> 5;    // M tile id (batch rows)
    const int n0      = blockIdx.x * 16;
    const int m0      = wave * 16;
    const int halfsel = lane >> 4;
    const int l16     = lane & 15;
    const int n       = n0 + l16;

    // ---- stage W_hh[n0..n0+15][0..1023] transposed into LDS, once ----
    {
        const int nl = threadIdx.x >> 4;          // 0..15
        const int kb = (threadIdx.x & 15) * 64;   // 64 consecutive k per thread
        const float* wrow = W_hh + (size_t)(n0 + nl) * H_ + kb;
        #pragma unroll
        for (int i = 0; i < 16; ++i) {
            v4f w = *(const v4f*)(wrow + i * 4);
            #pragma unroll
            for (int j = 0; j < 4; ++j) bsm[(kb + i * 4 + j) * 16 + nl] = w[j];
        }
    }
    __syncthreads();

    const float wo = W_out[n];
    // B fragment base: column l16, K offset 2*halfsel
    const float* bbase = bsm + l16 + 32 * halfsel;

    for (int t = 0; t < T_; ++t) {
        const float* hsrc = (t & 1) ? hB : hA;
        float*       hdst = (t & 1) ? hA : hB;

        // accumulator starts at the precomputed input projection for step t
        v8f c;
        #pragma unroll
        for (int v = 0; v < 8; ++v) {
            const int m = m0 + v + 8 * halfsel;          // batch index
            c[v] = xp[((size_t)m * T_ + t) * H_ + n];
        }

        const float* arow = hsrc + (size_t)(m0 + l16) * H_;

        #pragma unroll 8
        for (int kk = 0; kk < H_; kk += 4) {
            const int ka = kk + 2 * halfsel;
            v2f a = *(const v2f*)(arow + ka);
            v2f b;
            b[0] = bbase[kk * 16];
            b[1] = bbase[kk * 16 + 16];
            c = __builtin_amdgcn_wmma_f32_16x16x4_f32(
                false, a, false, b, (short)0, c, false, false);
        }

        // epilogue: tanh, write h_next, fused output projection
        #pragma unroll
        for (int v = 0; v < 8; ++v) {
            const int m = m0 + v + 8 * halfsel;
            float h = tanhf(c[v]);
            hdst[(size_t)m * H_ + n] = h;
            float p = h * wo;
            // reduce across the 16 lanes of each half-wave (same row)
            #pragma unroll
            for (int s = 1; s < 16; s <<= 1) p += __shfl_xor(p, s, 32);
            if (l16 == 0) atomicAdd(&out[(size_t)m * T_ + t], p);
        }

        // ---- device-wide split barrier (monotonic target: no reset races) --
        __syncthreads();
        __builtin_amdgcn_fence(__ATOMIC_RELEASE, "agent");   // publish h stores
        if (threadIdx.x == 0) {
            __hip_atomic_fetch_add(bar, 1u, __ATOMIC_RELEASE,
                                   __HIP_MEMORY_SCOPE_AGENT);
            const unsigned target = (unsigned)NBLK * (unsigned)(t + 1);
            while (__hip_atomic_load(bar, __ATOMIC_RELAXED,
                                     __HIP_MEMORY_SCOPE_AGENT) < target)
                __builtin_amdgcn_s_sleep(2);
        }
        __syncthreads();
        __builtin_amdgcn_fence(__ATOMIC_ACQUIRE, "agent");   // see h stores
    }
}

// ---------------------------------------------------------------------------
extern "C" void kernel_launch(void* const* d_in, const int* in_sizes, int n_in,
                              void* d_out, int out_size, void* d_ws, size_t ws_size,
                              hipStream_t stream) {
    (void)in_sizes; (void)n_in; (void)out_size; (void)ws_size;
    const float* x     = (const float*)d_in[0];
    const float* W_ih  = (const float*)d_in[1];
    const float* b_ih  = (const float*)d_in[2];
    const float* W_hh  = (const float*)d_in[3];
    const float* b_hh  = (const float*)d_in[4];
    const float* W_out = (const float*)d_in[5];
    const float* b_out = (const float*)d_in[6];
    float* out = (float*)d_out;

    // workspace: xp (256 MiB) | hA (512 KiB) | hB (512 KiB) | barrier counter
    char*  ws = (char*)d_ws;
    float* xp = (float*)ws;
    float* hA = (float*)(ws + (size_t)R_TOTAL * H_ * sizeof(float));
    float* hB = hA + B_ * H_;
    unsigned int* bar = (unsigned int*)(hB + B_ * H_);

    rnn_init <<<512, 256, 0, stream>>>(hA, out, b_out, bar);
    rnn_xproj<<<8192, 256, 0, stream>>>(x, W_ih, b_ih, b_hh, xp);
    rnn_recur<<<NBLK, 256, 0, stream>>>(xp, W_hh, W_out, hA, hB, out, bar);
}